// FeatMatchNN_63299228008958
// MI455X (gfx1250) — compile-verified
//
#include <hip/hip_runtime.h>

// ---------------------------------------------------------------------------
// FeatMatchNN forward for gfx1250 (MI455X).
// fp32 end-to-end (matches reference numerics). Decoder 3x3 convs are
// implicit-GEMM via V_WMMA_F32_16X16X4_F32: zero-padded im2col A-tile and
// weight tile staged in LDS, branch-free ds_load_b64 + wmma inner loop.
// Encoder convs / pool / cost volume are direct VALU kernels (3..16 channels
// starve the matrix pipe there).
// ---------------------------------------------------------------------------

typedef __attribute__((ext_vector_type(2))) float v2f;
typedef __attribute__((ext_vector_type(8))) float v8f;

#define NEG_SLOPE 0.01f

__device__ __forceinline__ float lrelu(float x) { return x >= 0.f ? x : NEG_SLOPE * x; }

// ---------------- encoder: direct 5x5 conv (pad 2) + bias + leaky relu -------
__global__ void conv5_lrelu(const float* __restrict__ in, const float* __restrict__ w,
                            const float* __restrict__ bias, float* __restrict__ out,
                            int N, int Cin, int Cout, int H, int W)
{
    int idx = blockIdx.x * blockDim.x + threadIdx.x;
    int total = N * Cout * H * W;
    if (idx >= total) return;
    int x = idx % W;
    int y = (idx / W) % H;
    int co = (idx / (W * H)) % Cout;
    int n = idx / (W * H * Cout);
    float acc = bias[co];
    const float* wp = w + (size_t)co * Cin * 25;
    for (int c = 0; c < Cin; ++c) {
        const float* ip = in + ((size_t)(n * Cin + c) * H) * W;
        for (int ky = 0; ky < 5; ++ky) {
            int yy = y + ky - 2;
            if (yy < 0 || yy >= H) continue;
            for (int kx = 0; kx < 5; ++kx) {
                int xx = x + kx - 2;
                if (xx < 0 || xx >= W) continue;
                acc += ip[(size_t)yy * W + xx] * wp[(c * 5 + ky) * 5 + kx];
            }
        }
    }
    out[idx] = lrelu(acc);
}

// ---------------- 2x2 max pool, stride 2 -------------------------------------
__global__ void pool2(const float* __restrict__ in, float* __restrict__ out,
                      int NC, int Ho, int Wo)
{
    int idx = blockIdx.x * blockDim.x + threadIdx.x;
    int total = NC * Ho * Wo;
    if (idx >= total) return;
    int x = idx % Wo;
    int y = (idx / Wo) % Ho;
    int nc = idx / (Wo * Ho);
    int Hi = Ho * 2, Wi = Wo * 2;
    const float* ip = in + (size_t)nc * Hi * Wi;
    float a = ip[(size_t)(2 * y) * Wi + 2 * x];
    float b = ip[(size_t)(2 * y) * Wi + 2 * x + 1];
    float c = ip[(size_t)(2 * y + 1) * Wi + 2 * x];
    float d = ip[(size_t)(2 * y + 1) * Wi + 2 * x + 1];
    out[idx] = fmaxf(fmaxf(a, b), fmaxf(c, d));
}

// ---------------- q[b,l,d] = kq_w @ h1pool[b,:,py,px] + kq_b -----------------
__global__ void compute_q(const float* __restrict__ h1p, const int* __restrict__ pts,
                          const float* __restrict__ qw, const float* __restrict__ qbias,
                          float* __restrict__ q, int B, int L, int C, int Hp, int Wp, int shift)
{
    int idx = blockIdx.x * blockDim.x + threadIdx.x;
    int D = 2 * C;
    int total = B * L * D;
    if (idx >= total) return;
    int d = idx % D;
    int l = (idx / D) % L;
    int b = idx / (D * L);
    int p0 = pts[(b * L + l) * 2 + 0] >> shift;  // row
    int p1 = pts[(b * L + l) * 2 + 1] >> shift;  // col
    float acc = qbias[d];
    for (int c = 0; c < C; ++c)
        acc += qw[d * C + c] * h1p[((size_t)(b * C + c) * Hp + p0) * Wp + p1];
    q[idx] = acc;
}

// ---------------- v[b,d,h,w] = kv_w @ h2pool[b,:,h,w] + kv_b -----------------
__global__ void compute_v(const float* __restrict__ h2p, const float* __restrict__ vw,
                          const float* __restrict__ vbias, float* __restrict__ v,
                          int B, int C, int Hp, int Wp)
{
    int idx = blockIdx.x * blockDim.x + threadIdx.x;
    int D = 2 * C;
    int total = B * D * Hp * Wp;
    if (idx >= total) return;
    int x = idx % Wp;
    int y = (idx / Wp) % Hp;
    int d = (idx / (Wp * Hp)) % D;
    int b = idx / (Wp * Hp * D);
    float acc = vbias[d];
    for (int c = 0; c < C; ++c)
        acc += vw[d * C + c] * h2p[((size_t)(b * C + c) * Hp + y) * Wp + x];
    v[idx] = acc;
}

// ---------------- attn[b*L+l,h,w] = sum_d q[b,l,d] * v[b,d,h,w] --------------
__global__ void compute_attn(const float* __restrict__ q, const float* __restrict__ v,
                             float* __restrict__ attn, int B, int L, int D, int Hp, int Wp)
{
    int idx = blockIdx.x * blockDim.x + threadIdx.x;
    int total = B * L * Hp * Wp;
    if (idx >= total) return;
    int x = idx % Wp;
    int y = (idx / Wp) % Hp;
    int l = (idx / (Wp * Hp)) % L;
    int b = idx / (Wp * Hp * L);
    float acc = 0.f;
    for (int d = 0; d < D; ++d)
        acc += q[(b * L + l) * D + d] * v[((size_t)(b * D + d) * Hp + y) * Wp + x];
    attn[((size_t)(b * L + l) * Hp + y) * Wp + x] = acc;
}

// ---------------- decoder input: concat(prev, attn) -> up2x -> concat(feats) -
// prev indexing: tiled (n % B, reference jnp.tile semantics) or direct (n).
// attn/output sample index n corresponds to (b = n/L, l = n%L).
__global__ void build_dec_input(const float* __restrict__ prev, const float* __restrict__ attn,
                                const float* __restrict__ feats, float* __restrict__ out,
                                int NL, int B, int Cprev, int Cf, int H, int W, int prev_tiled)
{
    int Ho = 2 * H, Wo = 2 * W;
    int Ccat = Cprev + 1 + Cf;
    size_t idx = (size_t)blockIdx.x * blockDim.x + threadIdx.x;
    size_t total = (size_t)NL * Ccat * Ho * Wo;
    if (idx >= total) return;
    int x = (int)(idx % Wo);
    int y = (int)((idx / Wo) % Ho);
    int c = (int)((idx / ((size_t)Wo * Ho)) % Ccat);
    int n = (int)(idx / ((size_t)Wo * Ho * Ccat));
    float val;
    if (c < Cprev) {
        int pn = prev_tiled ? (n % B) : n;
        val = prev[((size_t)(pn * Cprev + c) * H + (y >> 1)) * W + (x >> 1)];
    } else if (c == Cprev) {
        val = attn[((size_t)n * H + (y >> 1)) * W + (x >> 1)];
    } else {
        int fn = n % B;
        val = feats[((size_t)(fn * Cf + (c - Cprev - 1)) * Ho + y) * Wo + x];
    }
    out[idx] = val;
}

// ---------------- WMMA fp32 3x3 conv (pad 1), implicit GEMM ------------------
// One wave computes a 16(pixels) x 16(out channels) tile with K = Cin*9,
// padded to K16 = ceil(K/16)*16, chunked by 4 through V_WMMA_F32_16X16X4_F32.
//
// Fragment layout (CDNA5 ISA 7.12.2, 32-bit operands):
//   A 16x4  (2 VGPRs): lanes 0-15 M=lane K={k0,k0+1}; lanes 16-31 K={k0+2,k0+3}.
//   B 4x16  (2 VGPRs): mirrored (lane -> N column).
//   C/D 16x16 (8 VGPRs): VGPR r -> M = r + 8*(lane>=16), N = lane&15.
//
// A-tile is im2col'd into LDS (zero-padded rows/tail), weights zero-padded to
// a 16 x K16 tile, so the inner loop is two aligned b64 LDS pair-loads + wmma
// with no guards and no divergence (EXEC all-ones at every wmma).
#define CW_WAVES 2
#define CW_KP 304        // ceil(33*9/16)*16

__global__ __launch_bounds__(CW_WAVES * 32) void conv3_wmma(
    const float* __restrict__ in, const float* __restrict__ w,
    const float* __restrict__ bias, float* __restrict__ out,
    int Nb, int Cin, int Cout, int H, int W, int do_lrelu)
{
    __shared__ float s_w[16 * CW_KP];                 // [n][k], zero-padded
    __shared__ float s_a[CW_WAVES][16 * CW_KP];       // [m][k], zero-padded

    const int K = Cin * 9;
    const int K16 = (K + 15) & ~15;
    const int tid = threadIdx.x;
    const int lane = tid & 31;
    const int wave = tid >> 5;

    const int tilesPerRow = W >> 4;            // W is a multiple of 16 here
    const int tilesPerImg = H * tilesPerRow;
    const int totalTiles = Nb * tilesPerImg;
    const int tile = blockIdx.x * CW_WAVES + wave;
    const bool active = tile < totalTiles;

    int n = 0, y = 0, x0 = 0;
    if (active) {
        n = tile / tilesPerImg;
        int r = tile % tilesPerImg;
        y = r / tilesPerRow;
        x0 = (r % tilesPerRow) << 4;
        __builtin_prefetch(&in[((size_t)(n * Cin) * H + y) * W + x0], 0, 0);
    }

    // stage weights, zero-padded to 16 rows x K16 (no idiv: nested loops)
    for (int nrow = 0; nrow < 16; ++nrow)
        for (int k = tid; k < K16; k += CW_WAVES * 32) {
            float wv = 0.f;
            if (nrow < Cout && k < K) wv = w[nrow * K + k];
            s_w[nrow * K16 + k] = wv;
        }

    // build im2col A-tile: A[m][k] = in[n, c, y+ky-1, x0+m+kx-1] (0 outside)
    if (active) {
        float* ap = &s_a[wave][0];
        for (int m = 0; m < 16; ++m) {
            int gxbase = x0 + m - 1;
            for (int k = lane; k < K16; k += 32) {
                float val = 0.f;
                if (k < K) {
                    int c = k / 9;
                    int rr = k - c * 9;
                    int ky = rr / 3;
                    int kx = rr - ky * 3;
                    int gy = y + ky - 1;
                    int gx = gxbase + kx;
                    if (gy >= 0 && gy < H && gx >= 0 && gx < W)
                        val = in[((size_t)(n * Cin + c) * H + gy) * W + gx];
                }
                ap[m * K16 + k] = val;
            }
        }
    }
    __syncthreads();
    if (!active) return;

    const int half = lane >> 4;   // 0: K pair {k0,k0+1}; 1: {k0+2,k0+3}
    const int mn = lane & 15;     // A row (pixel) == B col (out channel)
    const float* arow = &s_a[wave][mn * K16 + 2 * half];
    const float* brow = &s_w[mn * K16 + 2 * half];

    v8f acc = {0.f, 0.f, 0.f, 0.f, 0.f, 0.f, 0.f, 0.f};
    for (int k0 = 0; k0 < K16; k0 += 16) {
#pragma unroll
        for (int u = 0; u < 4; ++u) {
            v2f a = *(const v2f*)(arow + k0 + 4 * u);   // ds_load_b64 (8B aligned)
            v2f b = *(const v2f*)(brow + k0 + 4 * u);
            acc = __builtin_amdgcn_wmma_f32_16x16x4_f32(
                /*neg_a=*/false, a, /*neg_b=*/false, b,
                /*c_mod=*/(short)0, acc, /*reuse_a=*/false, /*reuse_b=*/false);
        }
    }

    // D layout: per lane channel = mn, pixels x0 + half*8 + r (r = VGPR index)
    if (mn < Cout) {
        float bs = bias[mn];
        float* op = out + ((size_t)(n * Cout + mn) * H + y) * W + x0 + half * 8;
#pragma unroll
        for (int r = 0; r < 8; ++r) {
            float vv = acc[r] + bs;
            if (do_lrelu) vv = lrelu(vv);
            op[r] = vv;
        }
    }
}

// ---------------------------------------------------------------------------
extern "C" void kernel_launch(void* const* d_in, const int* in_sizes, int n_in,
                              void* d_out, int out_size, void* d_ws, size_t ws_size,
                              hipStream_t stream)
{
    (void)in_sizes; (void)n_in; (void)out_size; (void)ws_size;

    const float* x1 = (const float*)d_in[0];
    const float* x2 = (const float*)d_in[1];
    const int* pts = (const int*)d_in[2];
    const float *enc_w[4], *enc_b[4], *kq_w[4], *kq_b[4], *kv_w[4], *kv_b[4];
    for (int i = 0; i < 4; ++i) {
        enc_w[i] = (const float*)d_in[3 + 6 * i + 0];
        enc_b[i] = (const float*)d_in[3 + 6 * i + 1];
        kq_w[i]  = (const float*)d_in[3 + 6 * i + 2];
        kq_b[i]  = (const float*)d_in[3 + 6 * i + 3];
        kv_w[i]  = (const float*)d_in[3 + 6 * i + 4];
        kv_b[i]  = (const float*)d_in[3 + 6 * i + 5];
    }
    const float *dec_w[4], *dec_b[4];
    for (int i = 0; i < 4; ++i) {
        dec_w[i] = (const float*)d_in[27 + 2 * i];
        dec_b[i] = (const float*)d_in[28 + 2 * i];
    }
    const float* outc_w = (const float*)d_in[35];
    const float* outc_b = (const float*)d_in[36];

    const int B = 2, L = 16, NL = B * L;
    const int chan[5] = {3, 4, 8, 16, 16};
    const int S[4] = {384, 192, 96, 48};   // conv (pre-pool) resolution per level

    // --- workspace bump allocator -------------------------------------------
    char* ws = (char*)d_ws;
    size_t off = 0;
    auto alloc = [&](size_t nfloats) -> float* {
        float* p = (float*)(ws + off);
        off += ((nfloats * sizeof(float) + 255) & ~(size_t)255);
        return p;
    };
    float *h1c[4], *h2c[4], *h1p[4], *h2p[4], *vbuf[4], *qbuf[4], *attn[4];
    for (int i = 0; i < 4; ++i) {
        size_t conv_sz = (size_t)B * chan[i + 1] * S[i] * S[i];
        h1c[i] = alloc(conv_sz);
        h2c[i] = alloc(conv_sz);           // feats[i] (pre-pool features of x2)
        h1p[i] = alloc(conv_sz / 4);
        h2p[i] = alloc(conv_sz / 4);
        vbuf[i] = alloc((size_t)B * 2 * chan[i + 1] * (S[i] / 2) * (S[i] / 2));
        qbuf[i] = alloc((size_t)B * L * 2 * chan[i + 1]);
        attn[i] = alloc((size_t)NL * (S[i] / 2) * (S[i] / 2));
    }
    float* decin = alloc((size_t)NL * 9 * 384 * 384);   // largest decoder input
    float* decA  = alloc((size_t)NL * 4 * 192 * 192);   // outputs of steps 0, 2
    float* decB  = alloc((size_t)NL * 3 * 384 * 384);   // outputs of steps 1, 3

    const int TB = 256;
    auto nb = [&](size_t total) { return (int)((total + TB - 1) / TB); };

    // --- encoder + cost volumes ---------------------------------------------
    for (int i = 0; i < 4; ++i) {
        const float* in1 = (i == 0) ? x1 : h1p[i - 1];
        const float* in2 = (i == 0) ? x2 : h2p[i - 1];
        int Cin = chan[i], Cout = chan[i + 1], Hc = S[i];
        size_t ctot = (size_t)B * Cout * Hc * Hc;
        conv5_lrelu<<<nb(ctot), TB, 0, stream>>>(in1, enc_w[i], enc_b[i], h1c[i], B, Cin, Cout, Hc, Hc);
        conv5_lrelu<<<nb(ctot), TB, 0, stream>>>(in2, enc_w[i], enc_b[i], h2c[i], B, Cin, Cout, Hc, Hc);
        int Hp = Hc / 2;
        size_t ptot = (size_t)B * Cout * Hp * Hp;
        pool2<<<nb(ptot), TB, 0, stream>>>(h1c[i], h1p[i], B * Cout, Hp, Hp);
        pool2<<<nb(ptot), TB, 0, stream>>>(h2c[i], h2p[i], B * Cout, Hp, Hp);
        int D = 2 * Cout;
        size_t qtot = (size_t)B * L * D;
        compute_q<<<nb(qtot), TB, 0, stream>>>(h1p[i], pts, kq_w[i], kq_b[i], qbuf[i], B, L, Cout, Hp, Hp, i + 1);
        size_t vtot = (size_t)B * D * Hp * Hp;
        compute_v<<<nb(vtot), TB, 0, stream>>>(h2p[i], kv_w[i], kv_b[i], vbuf[i], B, Cout, Hp, Hp);
        size_t atot = (size_t)NL * Hp * Hp;
        compute_attn<<<nb(atot), TB, 0, stream>>>(qbuf[i], vbuf[i], attn[i], B, L, D, Hp, Hp);
    }

    // --- decoder -------------------------------------------------------------
    auto conv3 = [&](const float* cin, const float* cw, const float* cb, float* cout,
                     int Cin, int Cout, int Hc, int do_lr) {
        int tiles = NL * Hc * (Hc >> 4);
        int blocks = (tiles + CW_WAVES - 1) / CW_WAVES;
        conv3_wmma<<<blocks, CW_WAVES * 32, 0, stream>>>(cin, cw, cb, cout, NL, Cin, Cout, Hc, Hc, do_lr);
    };
    auto build = [&](const float* prev, const float* at, const float* ft, float* o,
                     int Cprev, int Cf, int Hh, int tiled) {
        size_t total = (size_t)NL * (Cprev + 1 + Cf) * (2 * Hh) * (2 * Hh);
        build_dec_input<<<nb(total), TB, 0, stream>>>(prev, at, ft, o, NL, B, Cprev, Cf, Hh, Hh, tiled);
    };

    // step 0: h = tile(h2p[3]) ++ attn3 -> up -> ++ feats3, conv 33->16 @48
    build(h2p[3], attn[3], h2c[3], decin, 16, 16, 24, /*tiled=*/1);
    conv3(decin, dec_w[0], dec_b[0], decA, 33, 16, 48, 1);
    // step 1: 33->8 @96
    build(decA, attn[2], h2c[2], decin, 16, 16, 48, 0);
    conv3(decin, dec_w[1], dec_b[1], decB, 33, 8, 96, 1);
    // step 2: 17->4 @192
    build(decB, attn[1], h2c[1], decin, 8, 8, 96, 0);
    conv3(decin, dec_w[2], dec_b[2], decA, 17, 4, 192, 1);
    // step 3: 9->3 @384
    build(decA, attn[0], h2c[0], decin, 4, 4, 192, 0);
    conv3(decin, dec_w[3], dec_b[3], decB, 9, 3, 384, 1);
    // output conv 3->1 @384, no activation; (B*L,1,H,W) == (B,L,H,W) row-major
    conv3(decB, outc_w, outc_b, (float*)d_out, 3, 1, 384, 0);
}